// AccuratePhysicsLoss_14035953123714
// MI455X (gfx1250) — compile-verified
//
#include <hip/hip_runtime.h>
#include <hip/hip_bf16.h>

// ---------------------------------------------------------------------------
// AccuratePhysicsLoss for MI455X (gfx1250)
//   f_now, f_next : (8, 4, 1024, 1024) fp32, NCHW contiguous
//   out           : 1 fp32 scalar
// HBM-bound: ~235 MB read -> ~10-12 us at 23.3 TB/s. One fused pass:
//   b128 async global->LDS staging of f_next (4-wide aligned halo) + f_now,
//   edge-correct torch.gradient 5-point stencils (interior tiles take a
//   branch-free specialized path), block partials to d_ws, then a wave32
//   v_wmma_f32_16x16x4_f32 reduction kernel folds 8192 partials + scale+clip.
// ---------------------------------------------------------------------------

#ifndef __has_builtin
#define __has_builtin(x) 0
#endif

#if __has_builtin(__builtin_amdgcn_global_load_async_to_lds_b128)
#define ASYNC_B128 1
#else
#define ASYNC_B128 0
#endif
#if __has_builtin(__builtin_amdgcn_global_load_async_to_lds_b32)
#define ASYNC_B32 1
#else
#define ASYNC_B32 0
#endif

#define W 1024
#define H 1024
#define BATCH 8
#define TX 64
#define TY 16
#define HX2 72                              // x halo of 4 each side (aligned)
#define HY (TY + 4)                         // y halo of 2 each side
#define TILES_X (W / TX)                    // 16
#define TILES_Y (H / TY)                    // 64
#define NBLOCKS (BATCH * TILES_X * TILES_Y) // 8192 partials (32 KB in d_ws)

typedef float v2f __attribute__((ext_vector_type(2)));
typedef float v8f __attribute__((ext_vector_type(8)));
typedef int v4i __attribute__((__vector_size__(4 * sizeof(int))));

typedef __attribute__((address_space(1))) int* gptr_t;
typedef __attribute__((address_space(3))) int* lptr_t;
typedef __attribute__((address_space(1))) v4i* gptr4_t;
typedef __attribute__((address_space(3))) v4i* lptr4_t;

// ---- CDNA5 async global -> LDS staging (16B chunks, both sides aligned) ----
__device__ __forceinline__ void stage_chunk(const float* g, float* l) {
#if ASYNC_B128
  __builtin_amdgcn_global_load_async_to_lds_b128((gptr4_t)g, (lptr4_t)l, 0, 0);
#elif ASYNC_B32
  __builtin_amdgcn_global_load_async_to_lds_b32((gptr_t)(g + 0), (lptr_t)(l + 0), 0, 0);
  __builtin_amdgcn_global_load_async_to_lds_b32((gptr_t)(g + 1), (lptr_t)(l + 1), 0, 0);
  __builtin_amdgcn_global_load_async_to_lds_b32((gptr_t)(g + 2), (lptr_t)(l + 2), 0, 0);
  __builtin_amdgcn_global_load_async_to_lds_b32((gptr_t)(g + 3), (lptr_t)(l + 3), 0, 0);
#else
  *(float4*)l = *(const float4*)g;
#endif
}

__device__ __forceinline__ void stage_wait() {
#if ASYNC_B128 || ASYNC_B32
#if __has_builtin(__builtin_amdgcn_s_wait_asynccnt)
  __builtin_amdgcn_s_wait_asynccnt(0);
#else
  asm volatile("s_wait_asynccnt 0" ::: "memory");
#endif
#endif
}

// ---- torch.gradient 1st+2nd derivative from 5 samples ---------------------
// d1 = tgrad(f)[g], d2 = tgrad(tgrad(f))[g]; n = axis length.
// EDGE=false: pure interior stencil (tile is >=2 away from every boundary).
// Clamped-halo phantom values are never referenced by the branch taken.
template <bool EDGE>
__device__ __forceinline__ void fd5(float m2, float m1, float c0, float p1,
                                    float p2, int g, int n, float& d1,
                                    float& d2) {
  if (!EDGE) {
    d1 = 0.5f * (p1 - m1);
    d2 = 0.25f * (p2 - 2.0f * c0 + m2);
    return;
  }
  if (g == 0) {
    d1 = p1 - c0;
    d2 = 0.5f * p2 - p1 + 0.5f * c0;
  } else if (g == 1) {
    d1 = 0.5f * (p1 - m1);
    d2 = 0.5f * m1 - 0.75f * c0 + 0.25f * p2;
  } else if (g == n - 1) {
    d1 = c0 - m1;
    d2 = 0.5f * c0 - m1 + 0.5f * m2;
  } else if (g == n - 2) {
    d1 = 0.5f * (p1 - m1);
    d2 = 0.5f * p1 - 0.75f * c0 + 0.25f * m2;
  } else {
    d1 = 0.5f * (p1 - m1);
    d2 = 0.25f * (p2 - 2.0f * c0 + m2);
  }
}

// ---- residual accumulation over this block's 16x64 tile --------------------
template <bool EDGE>
__device__ float compute_tile(const float* sN, const float* sO, int x0, int y0,
                              int tid) {
  const float* Ut = sN + 0 * HY * HX2;
  const float* Vt = sN + 1 * HY * HX2;
  const float* Tt = sN + 2 * HY * HX2;
  const float* Pt = sN + 3 * HY * HX2;

  float acc = 0.0f;
#pragma unroll
  for (int k = 0; k < (TY * TX) / 256; ++k) {
    const int p = k * 256 + tid;
    const int lyo = p >> 6;  // TX = 64
    const int lxo = p & 63;
    const int gx = x0 + lxo;
    const int gy = y0 + lyo;
    const int ci = (lyo + 2) * HX2 + (lxo + 4);

    const float u = Ut[ci], v = Vt[ci], t = Tt[ci];
    const float un = sO[0 * TY * TX + p];
    const float vn = sO[1 * TY * TX + p];
    const float tn = sO[2 * TY * TX + p];

    float udx, udxx, udy, udyy;
    fd5<EDGE>(Ut[ci - 2], Ut[ci - 1], u, Ut[ci + 1], Ut[ci + 2], gx, W, udx,
              udxx);
    fd5<EDGE>(Ut[ci - 2 * HX2], Ut[ci - HX2], u, Ut[ci + HX2], Ut[ci + 2 * HX2],
              gy, H, udy, udyy);
    float vdx, vdxx, vdy, vdyy;
    fd5<EDGE>(Vt[ci - 2], Vt[ci - 1], v, Vt[ci + 1], Vt[ci + 2], gx, W, vdx,
              vdxx);
    fd5<EDGE>(Vt[ci - 2 * HX2], Vt[ci - HX2], v, Vt[ci + HX2], Vt[ci + 2 * HX2],
              gy, H, vdy, vdyy);
    float tdx, tdxx, tdy, tdyy;
    fd5<EDGE>(Tt[ci - 2], Tt[ci - 1], t, Tt[ci + 1], Tt[ci + 2], gx, W, tdx,
              tdxx);
    fd5<EDGE>(Tt[ci - 2 * HX2], Tt[ci - HX2], t, Tt[ci + HX2], Tt[ci + 2 * HX2],
              gy, H, tdy, tdyy);
    float pdx, pdy, d2u; // second derivative of P unused
    fd5<EDGE>(Pt[ci - 2], Pt[ci - 1], Pt[ci], Pt[ci + 1], Pt[ci + 2], gx, W,
              pdx, d2u);
    fd5<EDGE>(Pt[ci - 2 * HX2], Pt[ci - HX2], Pt[ci], Pt[ci + HX2],
              Pt[ci + 2 * HX2], gy, H, pdy, d2u);

    // constants: 1/DT=100, PR=0.71, PR/DA=7.1, HA^2*PR=71, RA*PR=710,
    // diff_c = 1 + 4*RD/3 = 5/3, Q=0.1
    const float cont = udx + vdy;
    const float resx = (u - un) * 100.0f + u * udx + vn * udy + pdx -
                       0.71f * (udxx + udyy) + 7.1f * u;
    const float resy = (v - vn) * 100.0f + un * vdx + v * vdy + pdy -
                       0.71f * (vdxx + vdyy) - 710.0f * t + 78.1f * v;
    const float rest = (t - tn) * 100.0f + un * tdx + vn * tdy -
                       (5.0f / 3.0f) * (tdxx + tdyy) - 0.1f * t;

    acc += cont * cont + resx * resx + resy * resy + rest * rest;
  }
  return acc;
}

// ---------------------------------------------------------------------------
__global__ void __launch_bounds__(256) physics_tile_kernel(
    const float* __restrict__ f_now, const float* __restrict__ f_next,
    float* __restrict__ partials) {
  __shared__ alignas(16) float sN[4 * HY * HX2]; // f_next halo tiles (U,V,T,P)
  __shared__ alignas(16) float sO[3 * TY * TX];  // f_now tiles (U,V,T)
  __shared__ float sred[8];

  const int tid = threadIdx.x;
  const int x0 = blockIdx.x * TX;
  const int y0 = blockIdx.y * TY;
  const int b = blockIdx.z;

  const size_t bstride = (size_t)4 * H * W;
  const float* nb = f_next + (size_t)b * bstride;
  const float* ob = f_now + (size_t)b * bstride;

  // f_next: 4 ch x 20 rows x 18 aligned 4-float chunks (x range [x0-4,x0+68))
  // Chunk base clamped to [0, W-4]; clamped phantom slots are never consumed.
  for (int i = tid; i < 4 * HY * 18; i += 256) {
    const int c = i / (HY * 18);
    const int rem = i - c * (HY * 18);
    const int r = rem / 18;
    const int ch = rem - r * 18;
    int gy = y0 + r - 2;
    gy = gy < 0 ? 0 : (gy > H - 1 ? H - 1 : gy);
    int cgx = x0 - 4 + ch * 4;
    cgx = cgx < 0 ? 0 : (cgx > W - 4 ? W - 4 : cgx);
    stage_chunk(nb + ((size_t)c * H + gy) * W + cgx,
                &sN[(c * HY + r) * HX2 + ch * 4]);
  }
  // f_now: 3 ch x 16 rows x 16 chunks, no halo, always in-bounds/aligned
  for (int i = tid; i < 3 * TY * 16; i += 256) {
    const int c = i >> 8;         // / 256
    const int rem = i & 255;
    const int r = rem >> 4;
    const int ch = rem & 15;
    stage_chunk(ob + ((size_t)c * H + (y0 + r)) * W + x0 + ch * 4,
                &sO[(c * TY + r) * TX + ch * 4]);
  }
  stage_wait();
  __syncthreads();

  // ~85% of tiles touch no domain boundary -> branch-free stencils
  float acc;
  if (x0 != 0 && y0 != 0 && (x0 + TX) != W && (y0 + TY) != H)
    acc = compute_tile<false>(sN, sO, x0, y0, tid);
  else
    acc = compute_tile<true>(sN, sO, x0, y0, tid);

  // wave32 reduce, then block reduce
  float s = acc;
#pragma unroll
  for (int off = 16; off >= 1; off >>= 1) s += __shfl_xor(s, off, 32);
  if ((tid & 31) == 0) sred[tid >> 5] = s;
  __syncthreads();
  if (tid == 0) {
    float tot = 0.0f;
#pragma unroll
    for (int w = 0; w < 8; ++w) tot += sred[w];
    partials[((size_t)blockIdx.z * TILES_Y + blockIdx.y) * TILES_X +
             blockIdx.x] = tot;
  }
}

// ---------------------------------------------------------------------------
// WMMA-based final reduction: A = ones(16x4), B packs 64 partials (4x16);
// D += A*B accumulates column sums into every row of the f32 C tile, so
// sum(D over all 256 entries) == 16 * sum(all partials).
__global__ void __launch_bounds__(32) wmma_reduce_kernel(
    const float* __restrict__ partials, float* __restrict__ out) {
  const int lane = threadIdx.x;
  v2f a;
  a[0] = 1.0f;
  a[1] = 1.0f;
  v8f c = {};
#pragma unroll 4
  for (int it = 0; it < NBLOCKS / 64; ++it) {
    v2f bb;
    bb[0] = partials[it * 64 + lane * 2 + 0];
    bb[1] = partials[it * 64 + lane * 2 + 1];
    c = __builtin_amdgcn_wmma_f32_16x16x4_f32(false, a, false, bb, (short)0, c,
                                              false, false);
  }
  float s = 0.0f;
#pragma unroll
  for (int k = 0; k < 8; ++k) s += c[k];
#pragma unroll
  for (int off = 16; off >= 1; off >>= 1) s += __shfl_xor(s, off, 32);
  if (lane == 0) {
    // mean over 8*1024*1024 per residual, * BASE_SCALE(1e-4), then clip
    float total = (s * (1.0f / 16.0f)) * (1e-4f / 8388608.0f);
    total = fminf(fmaxf(total, 1e-10f), 1.0f);
    out[0] = total;
  }
}

// ---------------------------------------------------------------------------
extern "C" void kernel_launch(void* const* d_in, const int* in_sizes, int n_in,
                              void* d_out, int out_size, void* d_ws,
                              size_t ws_size, hipStream_t stream) {
  (void)in_sizes;
  (void)n_in;
  (void)out_size;
  (void)ws_size; // needs NBLOCKS*4 = 32 KB of d_ws
  const float* f_now = (const float*)d_in[0];
  const float* f_next = (const float*)d_in[1];
  float* partials = (float*)d_ws;
  float* out = (float*)d_out;

  dim3 grid(TILES_X, TILES_Y, BATCH);
  physics_tile_kernel<<<grid, 256, 0, stream>>>(f_now, f_next, partials);
  wmma_reduce_kernel<<<1, 32, 0, stream>>>(partials, out);
}